// maxpooler_ring_51393578664319
// MI455X (gfx1250) — compile-verified
//
#include <hip/hip_runtime.h>
#include <hip/hip_bf16.h>
#include <math.h>

// ---------------------------------------------------------------------------
// x:[B,DIN,N] f32, ring:[B,N] i32, W:[R,DOUT,DIN], b/gamma/beta/mean/var:[R,DOUT]
// out:[B,DOUT,N] f32
// out[b,o,n] = max_{m: ring[b,m]=ring[b,n]} (W'[r]·x[b,:,m])[o] + b'[r,o]
// ---------------------------------------------------------------------------
#define EPS 1e-5f
constexpr int B_   = 8;
constexpr int DIN  = 64;
constexpr int N_   = 16384;
constexpr int R_   = 4;
constexpr int DOUT = 128;

constexpr int NT        = 128;       // points per chunk in phase 1
constexpr int XS_STRIDE = NT + 16;   // LDS row pad: rows offset by 16 banks

// workspace layout (bytes)
constexpr size_t WPA_OFF = 0;        // v2f [R*8*16*32] = 131072 B  (swizzled A frags)
constexpr size_t BP_OFF  = 131072;   // float[R*DOUT]   = 2048 B    (folded bias)
constexpr size_t SEG_OFF = 133120;   // float[B*R*DOUT] = 16384 B   (segment max)

typedef __attribute__((ext_vector_type(2))) float v2f;
typedef __attribute__((ext_vector_type(8))) float v8f;
typedef __attribute__((ext_vector_type(4))) int   v4i;

#define AS_GLOBAL __attribute__((address_space(1)))
#define AS_LDS    __attribute__((address_space(3)))

#if defined(__has_builtin)
#if __has_builtin(__builtin_amdgcn_global_load_async_to_lds_b128)
#define HAVE_ASYNC_LDS 1
#endif
#endif

__device__ __forceinline__ void wait_async_done() {
#if defined(__has_builtin) && __has_builtin(__builtin_amdgcn_s_wait_asynccnt)
  __builtin_amdgcn_s_wait_asynccnt(0);
#else
  asm volatile("s_wait_asynccnt 0x0" ::: "memory");
#endif
}

// ---------------------------------------------------------------------------
// Kernel 0: fold BN into weights, pre-swizzle into WMMA A-fragment order,
//           write folded bias, init segmax to -inf (every launch).
// A-frag for V_WMMA_F32_16X16X4_F32 (A = 16Mx4K f32, 2 VGPR/lane):
//   lane L: M = L%16 ; V0 holds K = 2*(L/16), V1 holds K = 2*(L/16)+1
// ---------------------------------------------------------------------------
__global__ void prep_kernel(const float* __restrict__ W, const float* __restrict__ bconv,
                            const float* __restrict__ gamma, const float* __restrict__ beta,
                            const float* __restrict__ mean, const float* __restrict__ var,
                            v2f* __restrict__ WpA, float* __restrict__ bp,
                            float* __restrict__ seg) {
  int i = blockIdx.x * blockDim.x + threadIdx.x;   // i = r*DOUT + o, 0..511
  if (i >= R_ * DOUT) return;
  int r = i >> 7;
  int o = i & (DOUT - 1);
  float inv = gamma[i] * rsqrtf(var[i] + EPS);
  bp[i] = bconv[i] * inv + (beta[i] - mean[i] * inv);

  int ot = o >> 4;
  int lo = o & 15;
  const float* wrow = W + (size_t)i * DIN;
#pragma unroll
  for (int kk = 0; kk < 16; ++kk) {
#pragma unroll
    for (int hi = 0; hi < 2; ++hi) {
      int d0 = kk * 4 + 2 * hi;
      v2f v;
      v.x = wrow[d0] * inv;
      v.y = wrow[d0 + 1] * inv;
      WpA[((size_t)((r * 8 + ot) * 16 + kk)) * 32 + hi * 16 + lo] = v;
    }
  }
#pragma unroll
  for (int bb = 0; bb < B_; ++bb)
    seg[bb * (R_ * DOUT) + i] = -__builtin_inff();
}

// ---------------------------------------------------------------------------
// Kernel 1: ring-compacted GEMM-max via V_WMMA_F32_16X16X4_F32.
// grid = (N/NT, B), block = 256 (8 waves, wave w owns o-tile w for all rings).
// Points of each ring are compacted into 16-column tiles (padded with a
// duplicate index -> no masking needed; max is order/duplicate invariant).
// ---------------------------------------------------------------------------
__global__ __launch_bounds__(256) void segmax_kernel(
    const float* __restrict__ x, const int* __restrict__ ring,
    const v2f* __restrict__ WpA, float* __restrict__ seg) {
  __shared__ float Xs[DIN * XS_STRIDE];
  __shared__ int lists[R_][NT];
  __shared__ int cnt[R_];

  const int b   = blockIdx.y;
  const int n0  = blockIdx.x * NT;
  const int tid = threadIdx.x;

  if (tid < R_) cnt[tid] = 0;
  __syncthreads();

  // stage x[b, :, n0:n0+NT] -> Xs[d][n] (16B per thread per step)
  for (int idx = tid; idx < DIN * (NT / 4); idx += 256) {
    int d = idx >> 5;                 // NT/4 == 32
    int j = (idx & 31) * 4;
    const float* src = x + ((size_t)b * DIN + d) * N_ + n0 + j;
    float* dst = Xs + d * XS_STRIDE + j;
#if HAVE_ASYNC_LDS
    __builtin_amdgcn_global_load_async_to_lds_b128(
        (AS_GLOBAL v4i*)src, (AS_LDS v4i*)dst, 0, 0);
#else
    float4 v = *(const float4*)src;
    dst[0] = v.x; dst[1] = v.y; dst[2] = v.z; dst[3] = v.w;
#endif
  }
  // compact point indices by ring (order irrelevant for max)
  if (tid < NT) {
    int rv = ring[(size_t)b * N_ + n0 + tid];
    int slot = atomicAdd(&cnt[rv], 1);
    lists[rv][slot] = tid;
  }
  __syncthreads();
  // pad each ring's list to a multiple of 16 with a duplicate valid index
  if (tid < R_) {
    int c = cnt[tid];
    int e = (c + 15) & ~15;
    for (int s = c; s < e; ++s) lists[tid][s] = lists[tid][0];
  }
#if HAVE_ASYNC_LDS
  wait_async_done();
#endif
  __syncthreads();

  const int lane = tid & 31;
  const int hi   = lane >> 4;
  const int lo   = lane & 15;
  const int ot   = tid >> 5;          // o-tile == wave id (0..7)

  for (int r = 0; r < R_; ++r) {
    const int tiles = (cnt[r] + 15) >> 4;   // block-uniform -> EXEC stays full
    if (tiles == 0) continue;

    // coalesced A-fragment load, held in registers across tiles
    v2f a[16];
    const v2f* Ap = WpA + (size_t)((r * 8 + ot) * 16) * 32 + lane;
#pragma unroll
    for (int kk = 0; kk < 16; ++kk) a[kk] = Ap[(size_t)kk * 32];

    v8f cmax;
#pragma unroll
    for (int j = 0; j < 8; ++j) cmax[j] = -__builtin_inff();

    for (int t = 0; t < tiles; ++t) {
      const int col = lists[r][t * 16 + lo];   // this lane's point column
      v8f c = {0.f, 0.f, 0.f, 0.f, 0.f, 0.f, 0.f, 0.f};
#pragma unroll
      for (int kk = 0; kk < 16; ++kk) {
        // B frag (4Kx16N f32): V0: K=0 (lanes 0-15) / K=1 (lanes 16-31),
        //                      V1: K=2 / K=3
        const float* colp = Xs + (kk * 4 + hi) * XS_STRIDE + col;
        v2f bf;
        bf.x = colp[0];
        bf.y = colp[2 * XS_STRIDE];
        c = __builtin_amdgcn_wmma_f32_16x16x4_f32(
                false, a[kk], false, bf, (short)0, c, false, false);
      }
#pragma unroll
      for (int j = 0; j < 8; ++j) cmax[j] = fmaxf(cmax[j], c[j]);
    }

    // max across the 16 columns (within each 16-lane half)
#pragma unroll
    for (int m = 1; m < 16; m <<= 1) {
#pragma unroll
      for (int j = 0; j < 8; ++j) {
        float t2 = __shfl_xor(cmax[j], m, 32);
        cmax[j] = fmaxf(cmax[j], t2);
      }
    }
    if (lo == 0) {
#pragma unroll
      for (int j = 0; j < 8; ++j) {
        int oo = ot * 16 + j + 8 * hi;
        atomicMax(seg + ((size_t)b * R_ + r) * DOUT + oo, cmax[j]);
      }
    }
  }
}

// ---------------------------------------------------------------------------
// Kernel 2: out[b,o,n] = segmax[b, ring[b,n], o] + bias'[ring, o]
// grid = (N/1024, DOUT/32, B), block = 256; float4 coalesced stores (67 MB).
// ---------------------------------------------------------------------------
__global__ __launch_bounds__(256) void scatter_kernel(
    const int* __restrict__ ring, const float* __restrict__ seg,
    const float* __restrict__ bp, float* __restrict__ out) {
  __shared__ float segf[R_ * DOUT];
  const int b     = blockIdx.z;
  const int n0    = blockIdx.x * 1024;
  const int obase = blockIdx.y * 32;
  const int tid   = threadIdx.x;

  for (int i = tid; i < R_ * DOUT; i += 256)
    segf[i] = seg[b * (R_ * DOUT) + i] + bp[i];
  __syncthreads();

  const int n = n0 + tid * 4;
  const int4 rv = *(const int4*)(ring + (size_t)b * N_ + n);
#pragma unroll 4
  for (int oo = 0; oo < 32; ++oo) {
    int o = obase + oo;
    float s0 = segf[o];
    float s1 = segf[DOUT + o];
    float s2 = segf[2 * DOUT + o];
    float s3 = segf[3 * DOUT + o];
    float4 v;
    v.x = rv.x == 0 ? s0 : rv.x == 1 ? s1 : rv.x == 2 ? s2 : s3;
    v.y = rv.y == 0 ? s0 : rv.y == 1 ? s1 : rv.y == 2 ? s2 : s3;
    v.z = rv.z == 0 ? s0 : rv.z == 1 ? s1 : rv.z == 2 ? s2 : s3;
    v.w = rv.w == 0 ? s0 : rv.w == 1 ? s1 : rv.w == 2 ? s2 : s3;
    *(float4*)(out + ((size_t)b * DOUT + o) * N_ + n) = v;
  }
}

// ---------------------------------------------------------------------------
extern "C" void kernel_launch(void* const* d_in, const int* in_sizes, int n_in,
                              void* d_out, int out_size, void* d_ws, size_t ws_size,
                              hipStream_t stream) {
  (void)in_sizes; (void)n_in; (void)out_size; (void)ws_size;
  const float* x     = (const float*)d_in[0];
  const int*   ring  = (const int*)d_in[1];
  const float* W     = (const float*)d_in[2];
  const float* bconv = (const float*)d_in[3];
  const float* gamma = (const float*)d_in[4];
  const float* beta  = (const float*)d_in[5];
  const float* mean  = (const float*)d_in[6];
  const float* var   = (const float*)d_in[7];

  char*  ws  = (char*)d_ws;
  v2f*   WpA = (v2f*)(ws + WPA_OFF);
  float* bp  = (float*)(ws + BP_OFF);
  float* seg = (float*)(ws + SEG_OFF);
  float* out = (float*)d_out;

  prep_kernel<<<2, 256, 0, stream>>>(W, bconv, gamma, beta, mean, var, WpA, bp, seg);
  segmax_kernel<<<dim3(N_ / NT, B_), 256, 0, stream>>>(x, ring, WpA, seg);
  scatter_kernel<<<dim3(N_ / 1024, DOUT / 32, B_), 256, 0, stream>>>(ring, seg, bp, out);
}